// RadianceField_76854144795333
// MI455X (gfx1250) — compile-verified
//
#include <hip/hip_runtime.h>
#include <math.h>

// ---------------------------------------------------------------------------
// RadianceField trilinear sampling, MI455X (gfx1250, wave32).
//
// One 256-thread block (8 wave32s) per ray:
//   1) async-copy the ray's 256 samples global->LDS
//      (global_load_async_to_lds_b32 + s_wait_asynccnt, CDNA5 async path),
//   2) scale by scale_samples, bitonic-sort in LDS,
//   3) each thread handles one sorted sample: 8-corner gather of
//      (9 harmonics + 1 opacity) and trilinear blend, write 10 f32 outputs
//      with NON-TEMPORAL stores so the 604 MB grid keeps the 192 MB L2.
//
// Gather-bandwidth-bound workload (~0.1 GFLOP total vs ~1 GB of gather
// cacheline traffic -> tens of microseconds at 23.3 TB/s). The per-point
// 8x10 contraction has no shared matrix operand, so WMMA is not applicable.
// ---------------------------------------------------------------------------

#define NSAMP 256

#if __has_builtin(__builtin_amdgcn_global_load_async_to_lds_b32)
#define HAVE_ASYNC_COPY 1
#else
#define HAVE_ASYNC_COPY 0
#endif

typedef __attribute__((address_space(1))) int as1_int;  // global
typedef __attribute__((address_space(3))) int as3_int;  // LDS

__global__ __launch_bounds__(NSAMP) void rf_fused_kernel(
    const float* __restrict__ xs,       // (N,3) ray origins
    const float* __restrict__ dirs,     // (N,3) unit ray directions
    const float* __restrict__ samples,  // (N,S) uniform(0,1)
    const float* __restrict__ grid,     // (idim,idim,idim,9)
    const float* __restrict__ opacity,  // (idim,idim,idim)
    const int*   __restrict__ scale_p,  // scalar
    float* __restrict__ out_h,          // (N,S,9)
    float* __restrict__ out_o,          // (N,S)
    int idim)
{
    __shared__ float t_lds[NSAMP];

    const int ray = blockIdx.x;
    const int tid = threadIdx.x;
    const float scale = (float)(*scale_p);

    // ---- stage samples row into LDS (CDNA5 async global->LDS copy) ----
    const float* src = samples + (size_t)ray * NSAMP + tid;
#if HAVE_ASYNC_COPY
    __builtin_amdgcn_global_load_async_to_lds_b32(
        (as1_int*)src, (as3_int*)&t_lds[tid], /*offset=*/0, /*cpol=*/0);
#if __has_builtin(__builtin_amdgcn_s_wait_asynccnt)
    __builtin_amdgcn_s_wait_asynccnt(0);
#else
    asm volatile("s_wait_asynccnt 0" ::: "memory");
#endif
#else
    t_lds[tid] = *src;
#endif
    __syncthreads();

    // t = samples * scale (positive scale: sorting after is equivalent)
    t_lds[tid] *= scale;
    __syncthreads();

    // ---- bitonic sort of 256 values, 256 threads, LDS resident ----
#pragma unroll 1
    for (int k = 2; k <= NSAMP; k <<= 1) {
#pragma unroll 1
        for (int j = k >> 1; j > 0; j >>= 1) {
            const int ixj = tid ^ j;
            if (ixj > tid) {
                const float a = t_lds[tid];
                const float b = t_lds[ixj];
                const bool asc = ((tid & k) == 0);
                const bool do_swap = asc ? (a > b) : (a < b);
                if (do_swap) { t_lds[tid] = b; t_lds[ixj] = a; }
            }
            __syncthreads();
        }
    }

    const float t = t_lds[tid];

    // ---- sample point, voxel base, trilinear weights ----
    const float ox = xs[3 * ray + 0];
    const float oy = xs[3 * ray + 1];
    const float oz = xs[3 * ray + 2];
    const float dx = dirs[3 * ray + 0];
    const float dy = dirs[3 * ray + 1];
    const float dz = dirs[3 * ray + 2];

    const float px = ox + t * dx;
    const float py = oy + t * dy;
    const float pz = oz + t * dz;

    const float flx = floorf(px), fly = floorf(py), flz = floorf(pz);
    const float fx = px - flx, fy = py - fly, fz = pz - flz;

    // base is clamped; fractional part is NOT (matches reference exactly)
    int bx = (int)flx; bx = bx < 0 ? 0 : (bx > idim - 2 ? idim - 2 : bx);
    int by = (int)fly; by = by < 0 ? 0 : (by > idim - 2 ? idim - 2 : by);
    int bz = (int)flz; bz = bz < 0 ? 0 : (bz > idim - 2 ? idim - 2 : bz);

    const float wx[2] = {1.0f - fx, fx};
    const float wy[2] = {1.0f - fy, fy};
    const float wz0 = 1.0f - fz, wz1 = fz;

    float acc[9] = {0.f, 0.f, 0.f, 0.f, 0.f, 0.f, 0.f, 0.f, 0.f};
    float acco = 0.f;

    // 4 (x,y) corner pairs; the two z-neighbors are contiguous in memory:
    // 18 consecutive grid floats + 2 consecutive opacity floats per pair.
#pragma unroll
    for (int cx = 0; cx < 2; ++cx) {
#pragma unroll
        for (int cy = 0; cy < 2; ++cy) {
            const size_t vox0 =
                ((size_t)(bx + cx) * (size_t)idim + (size_t)(by + cy))
                * (size_t)idim + (size_t)bz;
            const float wxy = wx[cx] * wy[cy];
            const float w0 = wxy * wz0;
            const float w1 = wxy * wz1;
            const float* __restrict__ g = grid + vox0 * 9;  // 72B span, 4B aligned
#pragma unroll
            for (int h = 0; h < 9; ++h) acc[h] = fmaf(w0, g[h], acc[h]);
#pragma unroll
            for (int h = 0; h < 9; ++h) acc[h] = fmaf(w1, g[9 + h], acc[h]);
            const float* __restrict__ o = opacity + vox0;
            acco = fmaf(w0, o[0], acco);
            acco = fmaf(w1, o[1], acco);
        }
    }

    // ---- streamed output: non-temporal stores keep L2 for the grid ----
    const size_t pidx = (size_t)ray * NSAMP + tid;
    float* __restrict__ oh = out_h + pidx * 9;
#pragma unroll
    for (int h = 0; h < 9; ++h) __builtin_nontemporal_store(acc[h], oh + h);
    __builtin_nontemporal_store(acco, out_o + pidx);
}

extern "C" void kernel_launch(void* const* d_in, const int* in_sizes, int n_in,
                              void* d_out, int out_size, void* d_ws, size_t ws_size,
                              hipStream_t stream) {
    (void)n_in; (void)d_ws; (void)ws_size; (void)out_size;

    const float* xs      = (const float*)d_in[0];  // (N,3)
    const float* dirs    = (const float*)d_in[1];  // (N,3)
    const float* samples = (const float*)d_in[2];  // (N,S)
    const float* grid    = (const float*)d_in[3];  // (idim^3, 9)
    const float* opac    = (const float*)d_in[4];  // (idim^3)
    const int*   scale   = (const int*)d_in[5];    // scalar

    const int n_rays = in_sizes[0] / 3;
    const int n_samp = in_sizes[2] / n_rays;       // == 256 (NSAMP)

    // idim from opacity element count (idim^3)
    int idim = (int)llroundf(cbrtf((float)in_sizes[4]));
    while ((long long)idim * idim * idim < (long long)in_sizes[4]) ++idim;
    while ((long long)idim * idim * idim > (long long)in_sizes[4]) --idim;

    float* out_h = (float*)d_out;                                  // (N,S,9)
    float* out_o = out_h + (size_t)n_rays * (size_t)n_samp * 9;    // (N,S)

    rf_fused_kernel<<<n_rays, NSAMP, 0, stream>>>(
        xs, dirs, samples, grid, opac, scale, out_h, out_o, idim);
}